// MultiKeyframeProcessor_33749853012141
// MI455X (gfx1250) — compile-verified
//
#include <hip/hip_runtime.h>

typedef float v2f __attribute__((ext_vector_type(2)));
typedef float v4f __attribute__((ext_vector_type(4)));
typedef float v8f __attribute__((ext_vector_type(8)));

#define KFRAMES   5
#define T_FRAMES  257
#define HW        1024                         // 32*32
#define CHN       128
#define CHW       (CHN * HW)                   // 131072 per keyframe
#define OUT_LAT   ((size_t)CHN * T_FRAMES * HW)// 33,685,504 floats
#define NC_TILES  (CHW / 16)                   // 8192 chw tiles
#define CH_ITERS  8
#define STRIPS    (NC_TILES / CH_ITERS)        // 1024
#define T_TILES   17                           // ceil(257/16)

// Interpolation weight of keyframe k at frame t (reproduces reference exactly:
// before -> all weight on k=0, after -> all on k=K-1, at-keyframe -> 1 on j0,
// interior -> (1-alpha, alpha) on (j0, j1)). Returns 0 for padded k >= K.
__device__ __forceinline__ float kf_weight(int t, int k, const int fi[KFRAMES]) {
    int j = -1;
#pragma unroll
    for (int i = 0; i < KFRAMES; ++i)
        if (fi[i] <= t) j = i;
    const int j0 = (j < 0) ? 0 : j;
    int j1 = j + 1;
    if (j1 < 0) j1 = 0;
    if (j1 > KFRAMES - 1) j1 = KFRAMES - 1;

    const bool before   = (j < 0);
    const bool after    = (t > fi[KFRAMES - 1]);
    const bool at_kf    = (fi[j0] == t);
    const bool interior = !(before || after || at_kf);

    const float denom = fmaxf((float)(fi[j1] - fi[j0]), 1.0f);
    const float alpha = (float)(t - fi[j0]) / denom;
    const float w1 = interior ? alpha : 0.0f;
    const float w0 = 1.0f - w1;

    float w = 0.0f;
    if (k == j0) w += w0;
    if (k == j1) w += w1;
    return w;
}

// Out[CHW x T] = L[CHW x K(pad 8)] * W[K x T] via two chained f32 WMMAs per tile.
// Tile orientation: M = chw (rows), N = t (cols)  ->  each lane's 8 accumulator
// VGPRs are 8 consecutive chw elements at one t  ->  two b128 stores per lane.
__global__ void __launch_bounds__(256)
kf_lerp_wmma(const float* __restrict__ lat,
             const int*   __restrict__ fidx,
             float*       __restrict__ out)
{
    int fi[KFRAMES];
#pragma unroll
    for (int i = 0; i < KFRAMES; ++i) fi[i] = fidx[i];

    const int wave = blockIdx.x * (blockDim.x >> 5) + (threadIdx.x >> 5);
    const int lane = threadIdx.x & 31;
    const int half = lane >> 4;   // 0: lanes 0-15, 1: lanes 16-31
    const int lt   = lane & 15;

    const int tTile = wave / STRIPS;
    const int strip = wave % STRIPS;
    const int t0 = tTile * 16;
    const int t  = t0 + lt;       // this lane's output frame (N column)

    // B operand: 4x16 f32 (K x t-tile). VGPR0 = {K=0 | lanes0-15, K=2 | lanes16-31},
    // VGPR1 = {K=1, K=3}; second WMMA step covers K=4..7 (weights auto-zero for k>=5).
    v2f B0, B1;
    B0.x = kf_weight(t, half * 2 + 0, fi);
    B0.y = kf_weight(t, half * 2 + 1, fi);
    B1.x = kf_weight(t, 4 + half * 2 + 0, fi);
    B1.y = kf_weight(t, 4 + half * 2 + 1, fi);

    const int  kA0  = half * 2;       // 0 or 2 (step 0); 4 or 6 (step 1)
    const bool k4ok = (half == 0);    // only k=4 exists in step 1

#pragma unroll
    for (int it = 0; it < CH_ITERS; ++it) {
        const int tile  = strip * CH_ITERS + it;
        const int cbase = tile * 16;
        const int n     = cbase + lt;   // this lane's A row (chw index)

        // A operand: 16x4 f32. VGPR0 = {K=0 | lanes0-15, K=2 | lanes16-31}, VGPR1 = {K=1,K=3}.
        v2f A0, A1;
        A0.x = lat[(size_t)(kA0    ) * CHW + n];
        A0.y = lat[(size_t)(kA0 + 1) * CHW + n];
        A1.x = k4ok ? lat[(size_t)4 * CHW + n] : 0.0f;  // k=4 valid, k=6 padded 0
        A1.y = 0.0f;                                    // k=5 / k=7 padded 0

        v8f acc = {};
        acc = __builtin_amdgcn_wmma_f32_16x16x4_f32(false, A0, false, B0,
                                                    (short)0, acc, false, false);
        acc = __builtin_amdgcn_wmma_f32_16x16x4_f32(false, A1, false, B1,
                                                    (short)0, acc, false, false);

        if (t < T_FRAMES) {   // predicate only the store; EXEC full for WMMA above
            const int chw0 = cbase + half * 8;        // lane's first chw (M = r + 8*half)
            const int c    = chw0 >> 10;              // 16 | 1024, tile never crosses c
            const int hw   = chw0 & 1023;
            float* p = out + (size_t)c * ((size_t)T_FRAMES * HW)
                           + (size_t)t * HW + hw;
            v4f lo = { acc[0], acc[1], acc[2], acc[3] };
            v4f hi = { acc[4], acc[5], acc[6], acc[7] };
            // write-once stream: non-temporal so the 2.6MB latent set stays cache-hot
            __builtin_nontemporal_store(lo, reinterpret_cast<v4f*>(p));
            __builtin_nontemporal_store(hi, reinterpret_cast<v4f*>(p + 4));
        }
    }
}

__global__ void kf_mask(const float* __restrict__ s,
                        const int*   __restrict__ fidx,
                        float*       __restrict__ outm)
{
    const int t = blockIdx.x * blockDim.x + threadIdx.x;
    if (t >= T_FRAMES) return;

    int fi[KFRAMES];
#pragma unroll
    for (int i = 0; i < KFRAMES; ++i) fi[i] = fidx[i];

    int j = -1;
#pragma unroll
    for (int i = 0; i < KFRAMES; ++i)
        if (fi[i] <= t) j = i;
    const int j0 = (j < 0) ? 0 : j;
    int j1 = j + 1;
    if (j1 < 0) j1 = 0;
    if (j1 > KFRAMES - 1) j1 = KFRAMES - 1;

    const bool before   = (j < 0);
    const bool after    = (t > fi[KFRAMES - 1]);
    const bool at_kf    = (fi[j0] == t);
    const bool interior = !(before || after || at_kf);

    const float denom = fmaxf((float)(fi[j1] - fi[j0]), 1.0f);
    const float alpha = (float)(t - fi[j0]) / denom;
    const float w1 = interior ? alpha : 0.0f;
    const float w0 = 1.0f - w1;

    const float m_mid = w0 * s[j0] + w1 * s[j1];
    const float fi0 = (float)fi[0];
    const float fiL = (float)fi[KFRAMES - 1];
    const float decay_b = fmaxf(1.0f - (fi0 - (float)t) / fmaxf(fi0, 1.0f), 0.0f);
    const float decay_a = fmaxf(1.0f - ((float)t - fiL) / ((float)T_FRAMES - fiL), 0.0f);

    outm[t] = before ? s[0] * decay_b
            : (after ? s[KFRAMES - 1] * decay_a : m_mid);
}

extern "C" void kernel_launch(void* const* d_in, const int* in_sizes, int n_in,
                              void* d_out, int out_size, void* d_ws, size_t ws_size,
                              hipStream_t stream) {
    const float* lat  = (const float*)d_in[0];   // [K,1,128,32,32] f32
    const float* strn = (const float*)d_in[1];   // [K] f32
    const int*   fidx = (const int*)d_in[2];     // [K] i32
    float* out = (float*)d_out;                  // [1,128,257,32,32] ++ [1,257]

    const int waves  = T_TILES * STRIPS;         // 17408
    const int blocks = waves / 8;                // 2176 blocks of 8 waves
    hipLaunchKernelGGL(kf_lerp_wmma, dim3(blocks), dim3(256), 0, stream,
                       lat, fidx, out);
    hipLaunchKernelGGL(kf_mask, dim3(1), dim3(288), 0, stream,
                       strn, fidx, out + OUT_LAT);
}